// MemN2N_83193516523877
// MI455X (gfx1250) — compile-verified
//
#include <hip/hip_runtime.h>
#include <hip/hip_bf16.h>
#include <math.h>

typedef __attribute__((ext_vector_type(2))) float v2f;
typedef __attribute__((ext_vector_type(8))) float v8f;

// Problem constants (from reference setup_inputs)
constexpr int B    = 256;
constexpr int M    = 50;
constexpr int L    = 50;
constexpr int V    = 50257;
constexpr int D    = 128;
constexpr int HOPS = 3;

// pe[l][d] = 1 + 4*((d+1) - D/2)*((l+1) - L/2)/(D*L)   (reference position_encoding, transposed)
__device__ __forceinline__ float pos_enc(int l, int d) {
    return 1.0f + 4.0f * ((float)(d + 1) - (float)D * 0.5f)
                       * ((float)(l + 1) - (float)L * 0.5f)
                       / (float)D / (float)L;
}

// ---------------------------------------------------------------------------
// Kernel 1: u0[b][d] = sum_l emb[0][x_q[b][l]][d] * pe(l,d)
// grid = B, block = D (128 threads)
// ---------------------------------------------------------------------------
__global__ void query_embed_kernel(const int* __restrict__ x_q,
                                   const float* __restrict__ emb,
                                   float* __restrict__ u) {
    const int b = blockIdx.x;
    const int d = threadIdx.x;
    __shared__ int idx_s[L];
    if (d < L) idx_s[d] = x_q[b * L + d];
    __syncthreads();
    float acc = 0.0f;
    #pragma unroll 5
    for (int l = 0; l < L; ++l) {
        const size_t row = (size_t)idx_s[l] * D;
        acc += emb[row + d] * pos_enc(l, d);
    }
    u[b * D + d] = acc;
}

// ---------------------------------------------------------------------------
// Kernel 2: m[t][b][mm][d] = sum_l emb[t][x_e[b][mm][l]][d] * pe(l,d)
// grid = (B*M, 4 tables), block = D
// (All 4 tables = 103 MB -> L2-resident on the 192 MB L2; gathers after the
//  compulsory pass hit L2.)
// ---------------------------------------------------------------------------
__global__ void mem_embed_kernel(const int* __restrict__ x_e,
                                 const float* __restrict__ emb,
                                 float* __restrict__ mtab) {
    const int bm = blockIdx.x;          // b*M + mm
    const int t  = blockIdx.y;          // table 0..3
    const int d  = threadIdx.x;
    __shared__ int idx_s[L];
    if (d < L) idx_s[d] = x_e[(size_t)bm * L + d];
    __syncthreads();
    const float* tab = emb + (size_t)t * V * D;
    float acc = 0.0f;
    #pragma unroll 5
    for (int l = 0; l < L; ++l) {
        const size_t row = (size_t)idx_s[l] * D;
        acc += tab[row + d] * pos_enc(l, d);
    }
    mtab[((size_t)t * B * M + bm) * D + d] = acc;
}

// ---------------------------------------------------------------------------
// Kernel 3: all 3 hops for one batch element per block.
//   p = softmax(m_a . u); u += p . m_c
// grid = B, block = D (4 wave32s)
// ---------------------------------------------------------------------------
__global__ void hops_kernel(const float* __restrict__ mtab,
                            float* __restrict__ u_g) {
    const int b    = blockIdx.x;
    const int tid  = threadIdx.x;
    const int lane = tid & 31;
    const int wave = tid >> 5;

    __shared__ float u_s[D];
    __shared__ float sc[M];
    __shared__ float p[M];

    u_s[tid] = u_g[b * D + tid];
    __syncthreads();

    for (int hop = 0; hop < HOPS; ++hop) {
        const float* mA = mtab + ((size_t)hop       * B + b) * M * D;
        const float* mC = mtab + ((size_t)(hop + 1) * B + b) * M * D;

        // scores: one wave per memory row, 32-lane dot over D=128
        for (int mm = wave; mm < M; mm += 4) {
            float acc = 0.0f;
            #pragma unroll
            for (int d = lane; d < D; d += 32) acc += mA[mm * D + d] * u_s[d];
            #pragma unroll
            for (int off = 16; off > 0; off >>= 1) acc += __shfl_xor(acc, off, 32);
            if (lane == 0) sc[mm] = acc;
        }
        __syncthreads();

        // softmax over M=50 (cheap; single thread, deterministic)
        if (tid == 0) {
            float mx = -INFINITY;
            for (int mm = 0; mm < M; ++mm) mx = fmaxf(mx, sc[mm]);
            float s = 0.0f;
            for (int mm = 0; mm < M; ++mm) { float e = __expf(sc[mm] - mx); p[mm] = e; s += e; }
            const float inv = 1.0f / s;
            for (int mm = 0; mm < M; ++mm) p[mm] *= inv;
        }
        __syncthreads();

        // o[d] = sum_m p[m] * mC[m][d];  u += o
        float o = 0.0f;
        #pragma unroll 5
        for (int mm = 0; mm < M; ++mm) o += p[mm] * mC[mm * D + tid];
        __syncthreads();               // everyone done reading u_s for scores
        u_s[tid] += o;
        __syncthreads();
    }
    u_g[b * D + tid] = u_s[tid];
}

// ---------------------------------------------------------------------------
// Kernel 4: logits = u @ W via V_WMMA_F32_16X16X4_F32.
//
// M-blocked: each block owns 64 rows of u (4 M-tiles) x 128 columns
// (8 N-tiles, one per wave). Per K-step each wave loads its B-fragment ONCE
// and feeds 4 back-to-back WMMAs into independent accumulators (no D->A/B
// hazard), cutting W L2 traffic and B-side vmem instructions 4x vs a
// one-tile-per-wave scheme. W (25.7 MB) stays L2-resident.
//
// grid = (ceil(V/128)=393, B/64=4), block = 256 (8 waves).
// LDS u-tile 64x128 padded to stride 132 floats (conflict-free ds_load_b64
// A-fragment reads: lanes 0-15 banks 4j+k, lanes 16-31 banks 4j+k+2).
// ---------------------------------------------------------------------------
constexpr int U_STRIDE = 132;
constexpr int MTILES   = 4;     // 16-row tiles per block

__global__ void gemm_wmma_kernel(const float* __restrict__ u,
                                 const float* __restrict__ W,
                                 float* __restrict__ out) {
    const int tid  = threadIdx.x;
    const int lane = tid & 31;
    const int wave = tid >> 5;
    const int m0   = blockIdx.y * (16 * MTILES);
    const int n0   = blockIdx.x * 128 + wave * 16;

    __shared__ float u_s[16 * MTILES * U_STRIDE];

    // cooperative load of the 64xD u-tile
    for (int i = tid; i < 16 * MTILES * D; i += 256) {
        const int r = i >> 7;        // /128
        const int c = i & 127;
        u_s[r * U_STRIDE + c] = u[(m0 + r) * D + c];
    }
    __syncthreads();

    // A 16x4 fp32 fragment layout (ISA 7.12.2): lanes 0-15 hold M=lane,
    // VGPR0=K+0, VGPR1=K+1; lanes 16-31 hold M=lane-16, VGPR0=K+2, VGPR1=K+3.
    // B 4x16 row-striped: VGPR0 = rows K+0 / K+2, VGPR1 = rows K+1 / K+3.
    const int mrow  = lane & 15;
    const int khalf = (lane >> 4) << 1;         // 0 or 2
    const int ncol  = n0 + (lane & 15);
    const int ncolc = ncol < V ? ncol : V - 1;  // clamp loads: EXEC stays all-1s

    v8f c[MTILES];
    #pragma unroll
    for (int mt = 0; mt < MTILES; ++mt) c[mt] = (v8f){};

    for (int k = 0; k < D; k += 4) {
        v2f bfrag;
        bfrag.x = W[(size_t)(k + khalf)     * V + ncolc];
        bfrag.y = W[(size_t)(k + khalf + 1) * V + ncolc];
        if (k + 4 < D)
            __builtin_prefetch(&W[(size_t)(k + 4 + khalf) * V + ncolc], 0, 1);
        #pragma unroll
        for (int mt = 0; mt < MTILES; ++mt) {
            v2f a;
            a.x = u_s[(mt * 16 + mrow) * U_STRIDE + k + khalf];
            a.y = u_s[(mt * 16 + mrow) * U_STRIDE + k + khalf + 1];
            c[mt] = __builtin_amdgcn_wmma_f32_16x16x4_f32(
                        /*neg_a=*/false, a, /*neg_b=*/false, bfrag,
                        /*c_mod=*/(short)0, c[mt],
                        /*reuse_a=*/false, /*reuse_b=*/false);
        }
    }

    // C/D layout: VGPR r holds row base+r (lanes 0-15) / base+r+8 (lanes 16-31)
    if (ncol < V) {
        #pragma unroll
        for (int mt = 0; mt < MTILES; ++mt) {
            const int rbase = m0 + mt * 16 + ((lane >> 4) << 3);
            #pragma unroll
            for (int r = 0; r < 8; ++r)
                out[(size_t)(rbase + r) * V + ncol] = c[mt][r];
        }
    }
}

// ---------------------------------------------------------------------------
// Kernel 5: in-place row softmax over V. grid = B, block = 256.
// ---------------------------------------------------------------------------
__global__ void softmax_rows_kernel(float* __restrict__ out) {
    const int tid = threadIdx.x;
    float* row = out + (size_t)blockIdx.x * V;
    __shared__ float red[256];

    float mx = -INFINITY;
    for (int i = tid; i < V; i += 256) mx = fmaxf(mx, row[i]);
    red[tid] = mx;
    __syncthreads();
    for (int s = 128; s > 0; s >>= 1) {
        if (tid < s) red[tid] = fmaxf(red[tid], red[tid + s]);
        __syncthreads();
    }
    mx = red[0];
    __syncthreads();

    float sum = 0.0f;
    for (int i = tid; i < V; i += 256) {
        const float e = __expf(row[i] - mx);
        row[i] = e;
        sum += e;
    }
    red[tid] = sum;
    __syncthreads();
    for (int s = 128; s > 0; s >>= 1) {
        if (tid < s) red[tid] += red[tid + s];
        __syncthreads();
    }
    const float inv = 1.0f / red[0];
    for (int i = tid; i < V; i += 256) row[i] *= inv;
}

// ---------------------------------------------------------------------------
extern "C" void kernel_launch(void* const* d_in, const int* in_sizes, int n_in,
                              void* d_out, int out_size, void* d_ws, size_t ws_size,
                              hipStream_t stream) {
    const int*   x_e = (const int*)d_in[0];    // [B, M, L]
    const int*   x_q = (const int*)d_in[1];    // [B, L]
    const float* emb = (const float*)d_in[2];  // [4, V, D]
    const float* W   = (const float*)d_in[3];  // [D, V]
    float*       out = (float*)d_out;          // [B, V]

    // workspace: u [B*D], mtab [4*B*M*D]  (~26.4 MB)
    float* u    = (float*)d_ws;
    float* mtab = u + (size_t)B * D;

    query_embed_kernel<<<B, D, 0, stream>>>(x_q, emb, u);
    mem_embed_kernel<<<dim3(B * M, HOPS + 1), D, 0, stream>>>(x_e, emb, mtab);
    hops_kernel<<<B, D, 0, stream>>>(mtab, u);

    const int n_tiles = (V + 127) / 128;       // 393
    gemm_wmma_kernel<<<dim3(n_tiles, B / (16 * MTILES)), 256, 0, stream>>>(u, W, out);
    softmax_rows_kernel<<<B, 256, 0, stream>>>(out);
}